// GCNEncoder_55508157333634
// MI455X (gfx1250) — compile-verified
//
#include <hip/hip_runtime.h>

// GCN encoder for MI455X (gfx1250, wave32).
//
// h = relu(x@W_in^T + b_in)
// 3x: t = h@Wl^T ; h[n] = relu( dinv[n]^2*t[n] + sum_{e: dst=n} dinv[src]*dinv[n]*t[src] + b )
// out = relu(h@W_out^T + b_out)
//
// GEMMs: V_WMMA_F32_16X16X4_F32, 16x16 tile per wave, 32 wmma K-loop (verified
// pipelined codegen). Aggregation: CSR bucketing by dst built once per call
// (histogram + LDS scan + fill), then one wave per node accumulates in VGPRs
// with coalesced 512B gathers of t[src] -- zero f32 atomics in the hot loop.
// Feature tables (51MB each) and edge lists are L2-resident (192MB L2).

typedef float v2f __attribute__((ext_vector_type(2)));
typedef float v8f __attribute__((ext_vector_type(8)));

#define HID 128

// ---------------------------------------------------------------------------
// Y[m][n] = act( sum_k H[m][k] * W[n][k] (+ bias[n]) )   (Y = H @ W^T)
// Fragment layouts per CDNA5 ISA 7.12.2 (f32 16x16x4):
//   A: m = lane%16, k = vgpr + 2*(lane/16)
//   B: n = lane%16 (row of W), k = vgpr + 2*(lane/16)
//   C/D: n = lane%16, m = vgpr + 8*(lane/16)
// ---------------------------------------------------------------------------
__global__ __launch_bounds__(256) void gemm_nt_wmma(
    const float* __restrict__ H, const float* __restrict__ W,
    const float* __restrict__ bias, float* __restrict__ Y,
    int nrows, int relu_flag)
{
  const int lane = threadIdx.x & 31;
  const int wave = threadIdx.x >> 5;        // 0..7 -> output-column tile
  const int m0   = blockIdx.x * 16;
  const int n0   = wave * 16;
  const int lm   = lane & 15;
  const int half = lane >> 4;               // 0 or 1

  int arow = m0 + lm;
  if (arow >= nrows) arow = nrows - 1;      // clamp (EXEC must stay all-1 for WMMA)

  const float* __restrict__ hrow = H + (size_t)arow * HID + 2 * half;
  const float* __restrict__ wrow = W + (size_t)(n0 + lm) * HID + 2 * half;

  v8f acc = {0.f, 0.f, 0.f, 0.f, 0.f, 0.f, 0.f, 0.f};
#pragma unroll
  for (int k0 = 0; k0 < HID; k0 += 4) {
    v2f a = *(const v2f*)(hrow + k0);
    v2f b = *(const v2f*)(wrow + k0);
    acc = __builtin_amdgcn_wmma_f32_16x16x4_f32(false, a, false, b,
                                                (short)0, acc, false, false);
  }

  const float bn = bias ? bias[n0 + lm] : 0.0f;
#pragma unroll
  for (int v = 0; v < 8; ++v) {
    const int m = m0 + v + 8 * half;
    if (m < nrows) {
      float val = acc[v] + bn;
      if (relu_flag) val = fmaxf(val, 0.0f);
      Y[(size_t)m * HID + (n0 + lm)] = val;
    }
  }
}

// ---------------------------------------------------------------------------
// CSR construction by destination node
// ---------------------------------------------------------------------------
__global__ void cnt_zero(int* __restrict__ cnt, int N) {
  int i = blockIdx.x * blockDim.x + threadIdx.x;
  if (i < N) cnt[i] = 0;
}

__global__ void cnt_hist(const int* __restrict__ dst, int* __restrict__ cnt, int E) {
  int e = blockIdx.x * blockDim.x + threadIdx.x;
  if (e < E) atomicAdd(&cnt[dst[e]], 1);
}

// dinv[i] = rsqrt(deg) with deg = incoming-edge count + 1 (self loop)
__global__ void make_dinv(const int* __restrict__ cnt, float* __restrict__ dinv, int N) {
  int i = blockIdx.x * blockDim.x + threadIdx.x;
  if (i < N) dinv[i] = rsqrtf((float)cnt[i] + 1.0f);
}

// Exclusive prefix sum of cnt[0..N) -> offs[0..N). Single workgroup of 1024
// threads: serial chunk sums, LDS Hillis-Steele scan, serial chunk writeback.
#define SCAN_T 1024
__global__ __launch_bounds__(SCAN_T) void scan_excl(
    const int* __restrict__ cnt, int* __restrict__ offs, int N)
{
  __shared__ int part[SCAN_T];
  const int t = threadIdx.x;
  const int chunk = (N + SCAN_T - 1) / SCAN_T;
  const int beg = t * chunk;
  const int end = min(beg + chunk, N);

  int s = 0;
  for (int i = beg; i < end; ++i) s += cnt[i];
  part[t] = s;
  __syncthreads();

  for (int off = 1; off < SCAN_T; off <<= 1) {
    int v = (t >= off) ? part[t - off] : 0;
    __syncthreads();
    part[t] += v;
    __syncthreads();
  }

  int base = (t == 0) ? 0 : part[t - 1];
  for (int i = beg; i < end; ++i) { offs[i] = base; base += cnt[i]; }
}

// bucket[pos] = src[e] for pos carved out of dst[e]'s segment.
// offs is used as the live cursor: afterwards offs[n] == segment END.
__global__ void csr_fill(const int* __restrict__ src, const int* __restrict__ dst,
                         int* __restrict__ offs, int* __restrict__ bucket, int E)
{
  int e = blockIdx.x * blockDim.x + threadIdx.x;
  if (e < E) {
    int pos = atomicAdd(&offs[dst[e]], 1);
    bucket[pos] = src[e];
  }
}

// ---------------------------------------------------------------------------
// One wave per destination node:
//   h[n][:] = relu( dinv[n]^2 * t[n][:] + sum_s dinv[s]*dinv[n]*t[s][:] + b[:] )
// Lane L owns channels [4L, 4L+4): float4 accumulator in VGPRs, coalesced
// 512B/wave gathers of t rows, single plain store. No atomics.
// ---------------------------------------------------------------------------
__global__ __launch_bounds__(256) void gcn_aggregate(
    const float* __restrict__ t, const int* __restrict__ bucket,
    const int* __restrict__ segend, const int* __restrict__ cnt,
    const float* __restrict__ dinv, const float* __restrict__ bias,
    float* __restrict__ h, int N)
{
  const int node = (int)((blockIdx.x * blockDim.x + threadIdx.x) >> 5);
  const int lane = threadIdx.x & 31;
  if (node >= N) return;

  const int end = segend[node];          // offs[n] after fill = segment end
  const int beg = end - cnt[node];
  const float dn = dinv[node];
  const int c = lane * 4;

  // self-loop term (norm = dinv[n]^2)
  const float ws = dn * dn;
  float4 sv = *(const float4*)(t + (size_t)node * HID + c);
  float4 acc;
  acc.x = ws * sv.x; acc.y = ws * sv.y; acc.z = ws * sv.z; acc.w = ws * sv.w;

  int i = beg;
  for (; i + 2 <= end; i += 2) {         // 2-edge unroll to overlap loads
    const int s0 = bucket[i];
    const int s1 = bucket[i + 1];
    const float w0 = dinv[s0] * dn;
    const float w1 = dinv[s1] * dn;
    const float4 v0 = *(const float4*)(t + (size_t)s0 * HID + c);
    const float4 v1 = *(const float4*)(t + (size_t)s1 * HID + c);
    acc.x += w0 * v0.x + w1 * v1.x;
    acc.y += w0 * v0.y + w1 * v1.y;
    acc.z += w0 * v0.z + w1 * v1.z;
    acc.w += w0 * v0.w + w1 * v1.w;
  }
  if (i < end) {
    const int s0 = bucket[i];
    const float w0 = dinv[s0] * dn;
    const float4 v0 = *(const float4*)(t + (size_t)s0 * HID + c);
    acc.x += w0 * v0.x; acc.y += w0 * v0.y; acc.z += w0 * v0.z; acc.w += w0 * v0.w;
  }

  const float4 bv = *(const float4*)(bias + c);
  acc.x = fmaxf(acc.x + bv.x, 0.0f);
  acc.y = fmaxf(acc.y + bv.y, 0.0f);
  acc.z = fmaxf(acc.z + bv.z, 0.0f);
  acc.w = fmaxf(acc.w + bv.w, 0.0f);
  *(float4*)(h + (size_t)node * HID + c) = acc;
}

// ---------------------------------------------------------------------------
extern "C" void kernel_launch(void* const* d_in, const int* in_sizes, int n_in,
                              void* d_out, int out_size, void* d_ws, size_t ws_size,
                              hipStream_t stream)
{
  const float* x      = (const float*)d_in[0];   // [N,128]
  const int*   ei     = (const int*)  d_in[1];   // [2,E]
  const float* W_in   = (const float*)d_in[2];   // [128,128]
  const float* b_in   = (const float*)d_in[3];   // [128]
  const float* conv_W = (const float*)d_in[4];   // [3,128,128]
  const float* conv_b = (const float*)d_in[5];   // [3,128]
  const float* W_out  = (const float*)d_in[6];   // [128,128]
  const float* b_out  = (const float*)d_in[7];   // [128]

  const int N = in_sizes[0] / HID;
  const int E = in_sizes[1] / 2;
  const int* srcI = ei;
  const int* dstI = ei + E;

  // workspace: h | t | dinv | cnt | offs | bucket   (~117 MB)
  float* h      = (float*)d_ws;
  float* t      = h + (size_t)N * HID;
  float* dinv   = t + (size_t)N * HID;
  int*   cnt    = (int*)(dinv + N);
  int*   offs   = cnt + N;
  int*   bucket = offs + N;

  const int B = 256;
  const int gemm_blocks = (N + 15) / 16;
  const int node_blocks = (N + B - 1) / B;
  const int edge_blocks = (E + B - 1) / B;
  const int aggr_blocks = (N + 7) / 8;           // 8 waves (nodes) per block

  // CSR by dst + symmetric normalization (rebuilt every call; graph-replay safe)
  cnt_zero <<<node_blocks, B, 0, stream>>>(cnt, N);
  cnt_hist <<<edge_blocks, B, 0, stream>>>(dstI, cnt, E);
  make_dinv<<<node_blocks, B, 0, stream>>>(cnt, dinv, N);
  scan_excl<<<1, SCAN_T, 0, stream>>>(cnt, offs, N);
  csr_fill <<<edge_blocks, B, 0, stream>>>(srcI, dstI, offs, bucket, E);

  // fc_in + relu
  gemm_nt_wmma<<<gemm_blocks, B, 0, stream>>>(x, W_in, b_in, h, N, 1);

  // 3 GCN layers: dense transform (WMMA) + fused gather/aggregate/bias/relu
  for (int l = 0; l < 3; ++l) {
    gemm_nt_wmma <<<gemm_blocks, B, 0, stream>>>(h, conv_W + (size_t)l * HID * HID,
                                                 nullptr, t, N, 0);
    gcn_aggregate<<<aggr_blocks, B, 0, stream>>>(t, bucket, offs, cnt, dinv,
                                                 conv_b + (size_t)l * HID, h, N);
  }

  // fc_out + relu -> d_out
  gemm_nt_wmma<<<gemm_blocks, B, 0, stream>>>(h, W_out, b_out, (float*)d_out, N, 1);
}